// _RoITemporalPooling_5531917877637
// MI455X (gfx1250) — compile-verified
//
#include <hip/hip_runtime.h>
#include <hip/hip_bf16.h>
#include <stdint.h>

// Problem constants (from reference setup_inputs)
constexpr int C_ = 512, L_ = 96, H_ = 7, W_ = 7;
constexpr int HW_ = H_ * W_;          // 49
constexpr int SLICE = L_ * HW_;       // 4704 floats, contiguous per (n,c)
constexpr float NEGV = -1e30f;

#define AS1 __attribute__((address_space(1)))
#define AS3 __attribute__((address_space(3)))

typedef int v4i __attribute__((ext_vector_type(4)));

#if defined(__has_builtin)
#if __has_builtin(__builtin_amdgcn_global_load_async_to_lds_b128)
#define HAVE_ASYNC_B128_BUILTIN 1
#endif
#if __has_builtin(__builtin_amdgcn_s_wait_asynccnt)
#define HAVE_WAIT_ASYNC_BUILTIN 1
#endif
#endif

__device__ __forceinline__ void async_g2lds_16B(const float* gsrc, float* ldst) {
#if defined(HAVE_ASYNC_B128_BUILTIN)
    __builtin_amdgcn_global_load_async_to_lds_b128(
        (AS1 v4i*)gsrc, (AS3 v4i*)ldst, /*ioffset=*/0, /*cpol=*/0);
#else
    unsigned lds_addr = (unsigned)(uintptr_t)(AS3 void*)ldst;
    unsigned long long gaddr = (unsigned long long)(uintptr_t)gsrc;
    asm volatile("global_load_async_to_lds_b128 %0, %1, off"
                 :: "v"(lds_addr), "v"(gaddr) : "memory");
#endif
}

__device__ __forceinline__ void wait_async_zero() {
#if defined(HAVE_WAIT_ASYNC_BUILTIN)
    __builtin_amdgcn_s_wait_asynccnt(0);
#else
    asm volatile("s_wait_asynccnt 0" ::: "memory");
#endif
}

// Phase 1: spatial 2x2 overlapped max-pool.
// One block per (n,c). The (n,c) slice is 4704 contiguous floats -> DMA to LDS.
// Intermediate layout: [N, C, L, 4] where last dim = (ph*2+pw), stored as float4.
__global__ void spool_kernel(const float* __restrict__ feat,
                             float4* __restrict__ inter) {
    __shared__ float lds[SLICE];
    const int blk = blockIdx.x;                       // n*C + c
    const float* src = feat + (size_t)blk * SLICE;

    // Async DMA the whole slice into LDS (1176 x 16B, all 16B-aligned).
    for (int i = threadIdx.x; i < SLICE / 4; i += blockDim.x)
        async_g2lds_16B(src + i * 4, &lds[i * 4]);
    wait_async_zero();       // this wave's async writes to LDS are complete
    __syncthreads();         // make all waves' LDS data visible

    // bins: w0 = cols [0,4), w1 = cols [3,7); h0 = rows [0,4), h1 = rows [3,7)
    for (int l = threadIdx.x; l < L_; l += blockDim.x) {
        const float* row = &lds[l * HW_];
        float m00 = NEGV, m01 = NEGV, m10 = NEGV, m11 = NEGV;
#pragma unroll
        for (int h = 0; h < 7; ++h) {
            const float* r = row + h * 7;
            float a = fmaxf(fmaxf(r[0], r[1]), fmaxf(r[2], r[3]));
            float b = fmaxf(fmaxf(r[3], r[4]), fmaxf(r[5], r[6]));
            if (h < 4)  { m00 = fmaxf(m00, a); m01 = fmaxf(m01, b); }
            if (h >= 3) { m10 = fmaxf(m10, a); m11 = fmaxf(m11, b); }
        }
        inter[(size_t)blk * L_ + l] = make_float4(m00, m01, m10, m11);
    }
}

// Phase 2: per-ROI temporal max over 4 bins. One thread per (r,c).
// out layout [R, C, 4, 2, 2] -> float4 per (r,c,p).
__global__ void tpool_kernel(const float4* __restrict__ inter,
                             const float* __restrict__ rois,
                             float4* __restrict__ out, int R) {
    const int idx = blockIdx.x * blockDim.x + threadIdx.x;
    if (idx >= R * C_) return;
    const int r = idx / C_;
    const int c = idx - r * C_;

    const int   b  = (int)rois[r * 3 + 0];
    const float s  = rois[r * 3 + 1] * 0.125f;   // TEMPORAL_SCALE
    const float e  = rois[r * 3 + 2] * 0.125f;
    const float ctr  = (s + e) * 0.5f;
    const float half = (e - s) * 0.5f;           // CTX_RATIO = 1.0
    const float rs   = rintf(ctr - half);        // jnp.round == RNE
    const float re   = rintf(ctr + half);
    const float rlen = fmaxf(re - rs + 1.0f, 1.0f);
    const float binl = rlen * 0.25f;             // / POOLED_L

    const float4* base = inter + ((size_t)b * C_ + c) * L_;
    float4* dst = out + ((size_t)r * C_ + c) * 4;

#pragma unroll
    for (int p = 0; p < 4; ++p) {
        float lsf = fminf(fmaxf(floorf((float)p * binl) + rs, 0.0f), (float)L_);
        float lef = fminf(fmaxf(ceilf((float)(p + 1) * binl) + rs, 0.0f), (float)L_);
        int l0 = (int)lsf, l1 = (int)lef;
        float4 acc = make_float4(0.0f, 0.0f, 0.0f, 0.0f);
        if (l1 > l0) {                            // non-empty bin
            float4 m = make_float4(NEGV, NEGV, NEGV, NEGV);
            for (int l = l0; l < l1; ++l) {
                float4 v = base[l];
                m.x = fmaxf(m.x, v.x); m.y = fmaxf(m.y, v.y);
                m.z = fmaxf(m.z, v.z); m.w = fmaxf(m.w, v.w);
            }
            acc = m;
        }
        dst[p] = acc;                             // empty bin -> zeros
    }
}

extern "C" void kernel_launch(void* const* d_in, const int* in_sizes, int n_in,
                              void* d_out, int out_size, void* d_ws, size_t ws_size,
                              hipStream_t stream) {
    const float* feat = (const float*)d_in[0];
    const float* rois = (const float*)d_in[1];
    const int R = in_sizes[1] / 3;                       // 300
    const int N = in_sizes[0] / (C_ * L_ * H_ * W_);     // 2
    float4* inter = (float4*)d_ws;                       // N*C*L float4 = 1.5 MB

    spool_kernel<<<N * C_, 128, 0, stream>>>(feat, inter);

    const int total = R * C_;
    tpool_kernel<<<(total + 255) / 256, 256, 0, stream>>>(inter, rois,
                                                          (float4*)d_out, R);
}